// TransD_54846732370325
// MI455X (gfx1250) — compile-verified
//
#include <hip/hip_runtime.h>
#include <hip/hip_bf16.h>
#include <stdint.h>

// TransD scoring for MI455X (gfx1250, wave32).
// Memory-bound: ~218MB of random 512B entity-row gathers -> ~10us floor at
// 23.3 TB/s. Strategy: one wave per (batch, triple), 6 async B128 global->LDS
// row copies per wave (ASYNCcnt path, SADDR form with scalar row bases),
// wave32 butterfly reductions, no atomics.

#define HIDDEN  128
#define BATCH   4096
#define N_NEG   25
#define NPAIR   (N_NEG + 1)          // slot 0 = positive, 1..25 = negatives
#define MARGIN  4.0f
#define WAVES_PER_BLOCK 8
#define BLOCK   (WAVES_PER_BLOCK * 32)

__device__ __forceinline__ uint32_t lds_byte_addr(const void* p) {
  // Generic pointers to LDS carry the DS byte offset in the low 32 bits.
  return (uint32_t)(uintptr_t)p;
}

// CDNA5 async copy, GVS form: each lane copies 16B from global address sbase + voff to the LDS address in lds.
// Scalar 64-bit row base + one shared 32-bit lane offset -> minimal VGPRs.
__device__ __forceinline__ void async_row_b128(uint32_t lds, uint32_t voff,
                                               const float* sbase) {
  asm volatile("global_load_async_to_lds_b128 %0, %1, %2"
               :: "v"(lds), "v"(voff), "s"(sbase)
               : "memory");
}

__device__ __forceinline__ void wait_async0() {
#if __has_builtin(__builtin_amdgcn_s_wait_asynccnt)
  __builtin_amdgcn_s_wait_asynccnt(0);
#else
  asm volatile("s_wait_asynccnt 0" ::: "memory");
#endif
}

__device__ __forceinline__ float wave_sum(float v) {
#pragma unroll
  for (int m = 16; m >= 1; m >>= 1) v += __shfl_xor(v, m, 32);
  return v;
}

// ---------------------------------------------------------------------------
// Kernel A: one wave per (b, j) triple. scores[b*26 + j] = sum_H |h + r - t|.
// ---------------------------------------------------------------------------
__global__ __launch_bounds__(BLOCK) void transd_pair_scores(
    const float* __restrict__ ent_emb, const float* __restrict__ rel_emb,
    const float* __restrict__ ent_tr,  const float* __restrict__ rel_tr,
    const int* __restrict__ pos_h, const int* __restrict__ pos_t,
    const int* __restrict__ pos_r, const int* __restrict__ neg_h,
    const int* __restrict__ neg_t, const int* __restrict__ neg_r,
    float* __restrict__ scores)
{
  __shared__ float4 tiles[WAVES_PER_BLOCK][6][32];   // 24 KB / block

  const int w    = threadIdx.x >> 5;
  const int lane = threadIdx.x & 31;
  // Wave id is uniform: force it (and everything derived) onto the SALU path.
  const int wave =
      __builtin_amdgcn_readfirstlane(blockIdx.x * WAVES_PER_BLOCK + w);
  if (wave >= BATCH * NPAIR) return;

  const int b = wave / NPAIR;
  const int j = wave - b * NPAIR;

  int h_idx, t_idx, r_idx;
  if (j == 0) {
    h_idx = pos_h[b]; t_idx = pos_t[b]; r_idx = pos_r[b];
  } else {
    const int o = b * N_NEG + (j - 1);
    h_idx = neg_h[o]; t_idx = neg_t[o]; r_idx = neg_r[o];
  }
  h_idx = __builtin_amdgcn_readfirstlane(h_idx);
  t_idx = __builtin_amdgcn_readfirstlane(t_idx);
  r_idx = __builtin_amdgcn_readfirstlane(r_idx);

  // Scalar row bases (SGPR pairs), one shared per-lane byte offset.
  const float* ghe = ent_emb + ((size_t)h_idx << 7);
  const float* ght = ent_tr  + ((size_t)h_idx << 7);
  const float* gte = ent_emb + ((size_t)t_idx << 7);
  const float* gtt = ent_tr  + ((size_t)t_idx << 7);
  const float* gre = rel_emb + ((size_t)r_idx << 7);
  const float* grt = rel_tr  + ((size_t)r_idx << 7);

  const uint32_t voff = (uint32_t)(lane << 4);            // 16B per lane
  const uint32_t l0   = lds_byte_addr(&tiles[w][0][0]) + voff;
  async_row_b128(l0 + 0 * 512, voff, ghe);
  async_row_b128(l0 + 1 * 512, voff, ght);
  async_row_b128(l0 + 2 * 512, voff, gte);
  async_row_b128(l0 + 3 * 512, voff, gtt);
  async_row_b128(l0 + 4 * 512, voff, gre);
  async_row_b128(l0 + 5 * 512, voff, grt);
  wait_async0();   // wave-private LDS region: no barrier needed

  const float4 he = tiles[w][0][lane];
  const float4 ht = tiles[w][1][lane];
  const float4 te = tiles[w][2][lane];
  const float4 tt = tiles[w][3][lane];
  const float4 re = tiles[w][4][lane];
  const float4 rt = tiles[w][5][lane];

  // sc = (e_h . t_h) - (e_t . t_t): reduce the per-lane difference once.
  float d = (he.x * ht.x + he.y * ht.y + he.z * ht.z + he.w * ht.w)
          - (te.x * tt.x + te.y * tt.y + te.z * tt.z + te.w * tt.w);
  const float sc = wave_sum(d);

  float s = fabsf(he.x - te.x + sc * rt.x + re.x)
          + fabsf(he.y - te.y + sc * rt.y + re.y)
          + fabsf(he.z - te.z + sc * rt.z + re.z)
          + fabsf(he.w - te.w + sc * rt.w + re.w);
  s = wave_sum(s);

  if (lane == 0) scores[wave] = s;
}

// ---------------------------------------------------------------------------
// Kernel B: one wave per batch element: relu(p - mean(neg) + margin).
// 512 blocks -> one partial per block.
// ---------------------------------------------------------------------------
__global__ __launch_bounds__(BLOCK) void transd_batch_relu(
    const float* __restrict__ scores, float* __restrict__ partials)
{
  __shared__ float acc[WAVES_PER_BLOCK];
  const int w    = threadIdx.x >> 5;
  const int lane = threadIdx.x & 31;
  const int b    = blockIdx.x * WAVES_PER_BLOCK + w;

  float neg = 0.0f, pos = 0.0f;
  if (b < BATCH) {
    pos = scores[b * NPAIR];                               // broadcast load
    if (lane >= 1 && lane < NPAIR) neg = scores[b * NPAIR + lane];
  }
  neg = wave_sum(neg);
  float val = fmaxf(pos - neg * (1.0f / N_NEG) + MARGIN, 0.0f);
  if (b >= BATCH) val = 0.0f;

  if (lane == 0) acc[w] = val;
  __syncthreads();
  if (threadIdx.x == 0) {
    float t = 0.0f;
#pragma unroll
    for (int i = 0; i < WAVES_PER_BLOCK; ++i) t += acc[i];
    partials[blockIdx.x] = t;
  }
}

// ---------------------------------------------------------------------------
// Kernel C: single block folds the 512 partials into the scalar output.
// ---------------------------------------------------------------------------
__global__ __launch_bounds__(256) void transd_final_reduce(
    const float* __restrict__ partials, float* __restrict__ out, int n)
{
  __shared__ float sm[256];
  float v = 0.0f;
  for (int i = threadIdx.x; i < n; i += 256) v += partials[i];
  sm[threadIdx.x] = v;
  __syncthreads();
#pragma unroll
  for (int s = 128; s >= 1; s >>= 1) {
    if (threadIdx.x < s) sm[threadIdx.x] += sm[threadIdx.x + s];
    __syncthreads();
  }
  if (threadIdx.x == 0) out[0] = sm[0];
}

extern "C" void kernel_launch(void* const* d_in, const int* in_sizes, int n_in,
                              void* d_out, int out_size, void* d_ws, size_t ws_size,
                              hipStream_t stream) {
  const float* ent_emb = (const float*)d_in[0];
  const float* rel_emb = (const float*)d_in[1];
  const float* ent_tr  = (const float*)d_in[2];
  const float* rel_tr  = (const float*)d_in[3];
  const int*   pos_h   = (const int*)d_in[4];
  const int*   pos_t   = (const int*)d_in[5];
  const int*   pos_r   = (const int*)d_in[6];
  const int*   neg_h   = (const int*)d_in[7];
  const int*   neg_t   = (const int*)d_in[8];
  const int*   neg_r   = (const int*)d_in[9];

  float* scores   = (float*)d_ws;                 // BATCH*26 floats (~416 KB)
  float* partials = scores + BATCH * NPAIR;       // 512 floats

  const int total_waves = BATCH * NPAIR;                         // 106,496
  const int blocksA = (total_waves + WAVES_PER_BLOCK - 1) / WAVES_PER_BLOCK;
  const int blocksB = BATCH / WAVES_PER_BLOCK;                   // 512

  transd_pair_scores<<<blocksA, BLOCK, 0, stream>>>(
      ent_emb, rel_emb, ent_tr, rel_tr,
      pos_h, pos_t, pos_r, neg_h, neg_t, neg_r, scores);
  transd_batch_relu<<<blocksB, BLOCK, 0, stream>>>(scores, partials);
  transd_final_reduce<<<1, 256, 0, stream>>>(partials, (float*)d_out, blocksB);
}